// KCRouteEncoder_11261404250375
// MI455X (gfx1250) — compile-verified
//
#include <hip/hip_runtime.h>
#include <hip/hip_bf16.h>
#include <math.h>

typedef __attribute__((ext_vector_type(2))) float v2f;
typedef __attribute__((ext_vector_type(8))) float v8f;

#define BB   64
#define LL   200
#define KK   8
#define EMBD 256
#define PRE  768
#define NBL  (BB * LL)          // 12800 (row count, multiple of 128)

// ---------------------------------------------------------------------------
// Kernel 1: masked softmax over levels + pooling in both embedding spaces.
//   out_c[bl, 0:256]  = proj_b + sum_k alpha * cid_emb_padded[related]
//   pooledP[bl,0:768] = sum_k alpha * content_table[related]
// One 256-thread block per (b,l). Softmax over K=8 recomputed per thread.
// ---------------------------------------------------------------------------
__global__ void __launch_bounds__(256)
pool_kernel(const int*   __restrict__ croutes,       // [NBL, 8]
            const float* __restrict__ cid_emb,       // [NUM_C, 256]
            const float* __restrict__ weight,        // [8]
            const float* __restrict__ content_table, // [NUM_C+2, 768]
            const float* __restrict__ proj_b,        // [256]
            float*       __restrict__ out_c,         // [NBL, 256]
            float*       __restrict__ pooledP)       // [NBL, 768]
{
    const int bl = blockIdx.x;
    const int t  = threadIdx.x;

    const int* r = croutes + bl * KK;
    int   idx[KK];
    float w[KK], alpha[KK];
    float m = -INFINITY;
    #pragma unroll
    for (int k = 0; k < KK; ++k) {
        idx[k] = r[k] + 2;                 // related index into padded tables
        w[k]   = weight[k];
        if (idx[k] != 0 && w[k] > m) m = w[k];
    }
    float denom = 0.f;
    #pragma unroll
    for (int k = 0; k < KK; ++k) {
        float e = (idx[k] != 0) ? __expf(w[k] - m) : 0.f;
        alpha[k] = e;
        denom   += e;
    }
    const float inv = (denom > 0.f) ? (1.f / denom) : 0.f;
    #pragma unroll
    for (int k = 0; k < KK; ++k) alpha[k] *= inv;

    // pooled cid_emb (+ bias): rows 0/1 of the padded table are zeros -> skip.
    float acc = proj_b[t];
    #pragma unroll
    for (int k = 0; k < KK; ++k) {
        if (idx[k] >= 2 && alpha[k] != 0.f)
            acc += alpha[k] * cid_emb[(size_t)(idx[k] - 2) * EMBD + t];
    }
    out_c[(size_t)bl * EMBD + t] = acc;

    // pooled content_table rows (768 wide, 3 elements per thread).
    for (int c = t; c < PRE; c += 256) {
        float a = 0.f;
        #pragma unroll
        for (int k = 0; k < KK; ++k) {
            if (alpha[k] != 0.f)
                a += alpha[k] * content_table[(size_t)idx[k] * PRE + c];
        }
        pooledP[(size_t)bl * PRE + c] = a;
    }
}

// ---------------------------------------------------------------------------
// Kernel 2: fp32 GEMM with V_WMMA_F32_16X16X4_F32.
//   C[12800,256] += A[12800,768] @ Bw[768,256]     (C pre-seeded by kernel 1)
// Block = 256 threads = 8 wave32s. Block covers a 128-row x 16-col tile:
// the 768x16 B-panel is staged in LDS (48 KB), each wave owns one 16x16 tile.
// ---------------------------------------------------------------------------
__global__ void __launch_bounds__(256)
gemm_kernel(const float* __restrict__ A,   // [NBL, PRE]
            const float* __restrict__ Bw,  // [PRE, EMBD]
            float*       __restrict__ C)   // [NBL, EMBD] read C, write D
{
    __shared__ float Bs[PRE * 16];         // 48 KB

    const int n0     = blockIdx.x * 16;    // output column tile
    const int m_base = blockIdx.y * 128;   // output row super-tile

    // Cooperative stage of the 768x16 B panel.
    for (int f = threadIdx.x; f < PRE * 16; f += 256) {
        const int row = f >> 4;
        const int col = f & 15;
        Bs[f] = Bw[(size_t)row * EMBD + n0 + col];
    }
    __syncthreads();

    const int wave = threadIdx.x >> 5;
    const int lane = threadIdx.x & 31;
    const int m0   = m_base + wave * 16;
    const int half = lane >> 4;            // 0: K0/K1 side, 1: K2/K3 side
    const int ln   = lane & 15;

    // C/D layout: VGPR i holds row m0+i (lanes 0-15) / m0+8+i (lanes 16-31).
    const int crow = m0 + (half ? 8 : 0);
    v8f c;
    #pragma unroll
    for (int i = 0; i < 8; ++i)
        c[i] = C[(size_t)(crow + i) * EMBD + n0 + ln];

    // A layout (16x4 f32): lane ln holds row m0+ln; v0/v1 = K+2*half, K+2*half+1.
    const float* Arow = A + (size_t)(m0 + ln) * PRE;

    for (int k = 0; k < PRE; k += 4) {
        const int kk = k + 2 * half;
        v2f a;
        a.x = Arow[kk];
        a.y = Arow[kk + 1];
        v2f b;
        b.x = Bs[kk * 16 + ln];
        b.y = Bs[(kk + 1) * 16 + ln];
        // 8 args: (neg_a, A, neg_b, B, c_mod, C, reuse_a, reuse_b)
        c = __builtin_amdgcn_wmma_f32_16x16x4_f32(
                false, a, false, b, (short)0, c, false, false);
    }

    #pragma unroll
    for (int i = 0; i < 8; ++i)
        C[(size_t)(crow + i) * EMBD + n0 + ln] = c[i];
}

// ---------------------------------------------------------------------------
extern "C" void kernel_launch(void* const* d_in, const int* in_sizes, int n_in,
                              void* d_out, int out_size, void* d_ws, size_t ws_size,
                              hipStream_t stream)
{
    const int*   croutes       = (const int*)  d_in[0]; // [64,200,8]
    // d_in[1] = tailcs (unused by this branch of the reference)
    const float* cid_emb       = (const float*)d_in[2]; // [100000,256]
    const float* weight        = (const float*)d_in[3]; // [8]
    const float* content_table = (const float*)d_in[4]; // [100002,768]
    const float* proj_w        = (const float*)d_in[5]; // [768,256]
    const float* proj_b        = (const float*)d_in[6]; // [256]

    float* out     = (float*)d_out;                     // [12800,256]
    float* pooledP = (float*)d_ws;                      // [12800,768] = 39.3 MB

    pool_kernel<<<NBL, 256, 0, stream>>>(croutes, cid_emb, weight,
                                         content_table, proj_b, out, pooledP);

    dim3 grid(EMBD / 16, NBL / 128);                    // (16, 100)
    gemm_kernel<<<grid, 256, 0, stream>>>(pooledP, proj_w, out);
}